// FraudModel_65481071398833
// MI455X (gfx1250) — compile-verified
//
#include <hip/hip_runtime.h>

// FraudModel: x[B,2] -> 32 x (2x2 linear + tanh + scale/shift) -> 2->1 linear.
// Trans-unit (tanh) bound on MI455X: 1.07e9 tanh vs 201MB of streaming traffic.
// Strategy: per-lane VALU chain with native gfx1250 v_tanh_f32, scalar-path
// weight loads (uniform, 1.28KB total), non-temporal 128b loads / 64b stores.
// Block-uniform fast path: unconditional back-to-back loads (scalar branch).

constexpr int   LAYERS = 32;
constexpr int   BLOCK  = 256;   // 8 waves per workgroup (wave32)
constexpr int   PPT    = 4;     // row-pairs per thread -> 8 rows / thread
constexpr float CLIPV  = 5.0f;

typedef float v4f __attribute__((ext_vector_type(4)));
typedef float v2f __attribute__((ext_vector_type(2)));

__device__ __forceinline__ float tanh_hw(float v) {
#if __has_builtin(__builtin_amdgcn_tanhf)
    return __builtin_amdgcn_tanhf(v);          // lowers to v_tanh_f32 (gfx1250-new)
#else
    float r;
    asm("v_tanh_f32 %0, %1" : "=v"(r) : "v"(v));
    return r;
#endif
}

__device__ __forceinline__ float clip5(float v) {
    return fminf(fmaxf(v, -CLIPV), CLIPV);
}

// The 32-layer MLP chain on 2*PPT rows of register state.
__device__ __forceinline__ void mlp_chain(
    float* __restrict__ h0, float* __restrict__ h1,
    const float* __restrict__ W, const float* __restrict__ bias,
    const float* __restrict__ scale, const float* __restrict__ shift)
{
#pragma unroll
    for (int l = 0; l < LAYERS; ++l) {
        float w00 = W[4 * l + 0];
        float w01 = W[4 * l + 1];
        float w10 = W[4 * l + 2];
        float w11 = W[4 * l + 3];
        float b0  = bias[2 * l + 0];
        float b1  = bias[2 * l + 1];
        if (l != 0) {                 // layers 1..31 built with clip=True
            w00 = clip5(w00); w01 = clip5(w01);
            w10 = clip5(w10); w11 = clip5(w11);
            b0  = clip5(b0);  b1  = clip5(b1);
        }
        const float s0 = scale[2 * l + 0];
        const float s1 = scale[2 * l + 1];
        const float f0 = shift[2 * l + 0];
        const float f1 = shift[2 * l + 1];

#pragma unroll
        for (int r = 0; r < PPT * 2; ++r) {
            // h @ W^T + b  (torch Linear)
            const float a = fmaf(w00, h0[r], fmaf(w01, h1[r], b0));
            const float c = fmaf(w10, h0[r], fmaf(w11, h1[r], b1));
            // tanh -> * scale + shift, fused
            h0[r] = fmaf(tanh_hw(a), s0, f0);
            h1[r] = fmaf(tanh_hw(c), s1, f1);
        }
    }
}

__global__ __launch_bounds__(BLOCK, 1) void fraud_mlp_kernel(
    const float* __restrict__ x,
    const float* __restrict__ W,      // [L,2,2]
    const float* __restrict__ bias,   // [L,2]
    const float* __restrict__ scale,  // [L,2]
    const float* __restrict__ shift,  // [L,2]
    const float* __restrict__ W_out,  // [1,2]
    const float* __restrict__ b_out,  // [1]
    float* __restrict__ out,          // [B,1]
    int npairs)
{
    const int base = blockIdx.x * (BLOCK * PPT) + threadIdx.x;

    // 8 rows of state per thread: (h0,h1) per row.
    float h0[PPT * 2];
    float h1[PPT * 2];

    // BLOCK-UNIFORM (scalar) fast/tail selection: depends only on blockIdx.x,
    // so it lowers to s_cmp + s_cbranch, not an exec-mask dance. For the
    // reference shape (npairs = 8192 * 1024) every block is "full".
    const bool full = (int)((blockIdx.x + 1) * (BLOCK * PPT)) <= npairs;

    if (full) {
        // All 4 NT 128b loads issue unconditionally back-to-back
        // (each is a coalesced 512B wave transaction).
#pragma unroll
        for (int k = 0; k < PPT; ++k) {
            const v4f v = __builtin_nontemporal_load(
                (const v4f*)(x + 4 * (long long)(base + k * BLOCK)));
            h0[2 * k + 0] = v.x;  h1[2 * k + 0] = v.y;   // row 2p
            h0[2 * k + 1] = v.z;  h1[2 * k + 1] = v.w;   // row 2p+1
        }
    } else {
#pragma unroll
        for (int k = 0; k < PPT; ++k) {
            const int p = base + k * BLOCK;
            v4f v = {0.0f, 0.0f, 0.0f, 0.0f};
            if (p < npairs)
                v = __builtin_nontemporal_load((const v4f*)(x + 4 * (long long)p));
            h0[2 * k + 0] = v.x;  h1[2 * k + 0] = v.y;
            h0[2 * k + 1] = v.z;  h1[2 * k + 1] = v.w;
        }
    }

    mlp_chain(h0, h1, W, bias, scale, shift);

    const float wo0 = W_out[0];
    const float wo1 = W_out[1];
    const float bo  = b_out[0];

    // One float per row; a pair packs into a coalesced 8B NT store.
    if (full) {
#pragma unroll
        for (int k = 0; k < PPT; ++k) {
            v2f o;
            o.x = fmaf(wo0, h0[2 * k + 0], fmaf(wo1, h1[2 * k + 0], bo));
            o.y = fmaf(wo0, h0[2 * k + 1], fmaf(wo1, h1[2 * k + 1], bo));
            __builtin_nontemporal_store(
                o, (v2f*)(out + 2 * (long long)(base + k * BLOCK)));
        }
    } else {
#pragma unroll
        for (int k = 0; k < PPT; ++k) {
            const int p = base + k * BLOCK;
            if (p < npairs) {
                v2f o;
                o.x = fmaf(wo0, h0[2 * k + 0], fmaf(wo1, h1[2 * k + 0], bo));
                o.y = fmaf(wo0, h0[2 * k + 1], fmaf(wo1, h1[2 * k + 1], bo));
                __builtin_nontemporal_store(o, (v2f*)(out + 2 * (long long)p));
            }
        }
    }
}

extern "C" void kernel_launch(void* const* d_in, const int* in_sizes, int n_in,
                              void* d_out, int out_size, void* d_ws, size_t ws_size,
                              hipStream_t stream) {
    const float* x     = (const float*)d_in[0];
    const float* W     = (const float*)d_in[1];
    const float* bias  = (const float*)d_in[2];
    const float* scale = (const float*)d_in[3];
    const float* shift = (const float*)d_in[4];
    const float* W_out = (const float*)d_in[5];
    const float* b_out = (const float*)d_in[6];
    float*       out   = (float*)d_out;

    const int B      = in_sizes[0] / 2;      // x is [B,2]
    const int npairs = (B + 1) / 2;          // B = 16,777,216 -> 8,388,608 pairs
    const int pairs_per_block = BLOCK * PPT; // 1024
    const int grid = (npairs + pairs_per_block - 1) / pairs_per_block;  // 8192

    fraud_mlp_kernel<<<grid, BLOCK, 0, stream>>>(x, W, bias, scale, shift,
                                                 W_out, b_out, out, npairs);
}